// CGNN_85220741087437
// MI455X (gfx1250) — compile-verified
//
#include <hip/hip_runtime.h>
#include <hip/hip_bf16.h>
#include <cstdint>

typedef float v2f __attribute__((ext_vector_type(2)));
typedef float v8f __attribute__((ext_vector_type(8)));
typedef unsigned int u32x4 __attribute__((ext_vector_type(4)));
typedef int i32x4 __attribute__((ext_vector_type(4)));
typedef int i32x8 __attribute__((ext_vector_type(8)));

#if defined(__has_builtin)
#if __has_builtin(__builtin_amdgcn_tensor_load_to_lds) && \
    __has_builtin(__builtin_amdgcn_s_wait_tensorcnt)
#define HAVE_TDM 1
#else
#define HAVE_TDM 0
#endif
#else
#define HAVE_TDM 0
#endif

namespace {
constexpr int B_DIM = 32;
constexpr int N_DIM = 1024;
constexpr int D_DIM = 256;
constexpr int STEPS = 8;
constexpr float DT = 1.0f / 8.0f;   // SPATIAL_SCALE / N_STEPS

constexpr int KCH    = 32;          // k-chunk rows per TDM tile (diffusion GEMM)
constexpr int NCHUNK = N_DIM / KCH; // 32 pipeline stages
constexpr int SH_S   = D_DIM + 4;   // 260: padded LDS stride (TDM pad: 4 dw / 256 dw)
constexpr int SADJ_S = KCH + 1;     // 33 : padded LDS stride (TDM pad: 1 dw / 32 dw)
constexpr int SROW_S = D_DIM + 1;   // 257: stride for fc/weff row tiles
}

static __device__ __forceinline__ v8f wmma_f32_16x16x4(v2f a, v2f b, v8f c) {
  // D = A(16x4,f32) * B(4x16,f32) + C(16x16,f32), wave32
  return __builtin_amdgcn_wmma_f32_16x16x4_f32(false, a, false, b, (short)0, c,
                                               false, false);
}

// ---------------------------------------------------------------------------
// Tensor Data Mover: 2D tile (rows x cols of f32) global -> LDS, with LDS
// row padding applied by the TDM itself (D# per CDNA5 ISA).
//   group0: [1:0]=count=1, [63:32]=lds_addr, [120:64]=global_addr, [127:126]=2
//   group1: data_size=4B(2), pad_enable, pad_interval/amount codes,
//           tensor_dim0/1, tile_dim0/1, tensor_dim0_stride
// ---------------------------------------------------------------------------
#if HAVE_TDM
static __device__ __forceinline__ void tdm_load_2d_f32(
    unsigned lds_byte_off, const float* gaddr, unsigned tile_cols,
    unsigned tile_rows, unsigned tensor_cols, unsigned tensor_rows,
    unsigned row_stride_elems, unsigned pad_interval_code,
    unsigned pad_amount_code) {
  const unsigned long long ga = (unsigned long long)(uintptr_t)gaddr;
  u32x4 g0;
  g0.x = 1u;                                             // count=1 (user D#)
  g0.y = lds_byte_off;                                   // lds_addr (bytes)
  g0.z = (unsigned)(ga & 0xFFFFFFFFull);                 // global_addr[31:0]
  g0.w = (unsigned)((ga >> 32) & 0x01FFFFFFu) | (2u << 30);  // addr[56:32]|type=2
  i32x8 g1;
  g1[0] = (int)((2u << 16) |                 // data_size = 4B
                (1u << 20) |                 // pad_enable
                (pad_interval_code << 22) |  // dwords between pads
                (pad_amount_code << 25));    // pad dwords
  g1[1] = (int)((tensor_cols & 0xFFFFu) << 16);                       // dim0 lo
  g1[2] = (int)(((tensor_cols >> 16) & 0xFFFFu) |
                ((tensor_rows & 0xFFFFu) << 16));                     // dim0 hi|dim1 lo
  g1[3] = (int)(((tensor_rows >> 16) & 0xFFFFu) |
                ((tile_cols & 0xFFFFu) << 16));                       // dim1 hi|tile0
  g1[4] = (int)(tile_rows & 0xFFFFu);                                 // tile1 (tile2=0)
  g1[5] = (int)row_stride_elems;                                      // dim0_stride lo
  g1[6] = 0;
  g1[7] = 0;
  const i32x4 z4 = {0, 0, 0, 0};
#if __clang_major__ >= 23
  const i32x8 z8 = {0, 0, 0, 0, 0, 0, 0, 0};
  __builtin_amdgcn_tensor_load_to_lds(g0, g1, z4, z4, z8, 0);
#else
  __builtin_amdgcn_tensor_load_to_lds(g0, g1, z4, z4, 0);
#endif
}
#endif

// ---------------------------------------------------------------------------
// Kernel 1: h0[m, e] = sum_k x[m, k] * fc_in_w[e, k] + fc_in_b[e]
// ---------------------------------------------------------------------------
__global__ void __launch_bounds__(256)
fcin_kernel(const float* __restrict__ x, const float* __restrict__ wfc,
            const float* __restrict__ bfc, float* __restrict__ hout)
{
  const int m0   = blockIdx.x * 16;
  const int d0   = blockIdx.y * 128;
  const int tid  = threadIdx.x;
  const int wave = tid >> 5;
  const int lane = tid & 31;
  const int dw   = d0 + wave * 16;
  const int arow = lane & 15;
  const int koff = (lane >> 4) * 2;
  const int col  = lane & 15;

  __shared__ float sX[16 * SROW_S];
  for (int i = tid; i < 16 * D_DIM; i += 256) {
    const int r = i >> 8, k = i & (D_DIM - 1);
    sX[r * SROW_S + k] = x[(size_t)(m0 + r) * D_DIM + k];
  }
  __syncthreads();

  v8f acc = {0.f, 0.f, 0.f, 0.f, 0.f, 0.f, 0.f, 0.f};
  #pragma unroll 4
  for (int k = 0; k < D_DIM; k += 4) {
    v2f a, b;
    a.x = sX[arow * SROW_S + k + koff];
    a.y = sX[arow * SROW_S + k + koff + 1];
    b.x = wfc[(size_t)(dw + col) * D_DIM + k + koff];       // B[k][e]=wfc[e][k]
    b.y = wfc[(size_t)(dw + col) * D_DIM + k + koff + 1];
    acc = wmma_f32_16x16x4(a, b, acc);
  }

  const int rbase = (lane >> 4) * 8;
  const float bias = bfc[dw + col];
  #pragma unroll
  for (int v = 0; v < 8; ++v) {
    hout[(size_t)(m0 + rbase + v) * D_DIM + dw + col] = acc[v] + bias;
  }
}

// ---------------------------------------------------------------------------
// Kernel 2: w_eff[i, j] = sum_k w[i, k] * clip(d[k],0,1) * w[j, k]
// ---------------------------------------------------------------------------
__global__ void __launch_bounds__(256)
weff_kernel(const float* __restrict__ w, const float* __restrict__ dvec,
            float* __restrict__ weff)
{
  const int i0   = blockIdx.x * 16;
  const int j0   = blockIdx.y * 128;
  const int tid  = threadIdx.x;
  const int wave = tid >> 5;
  const int lane = tid & 31;
  const int jw   = j0 + wave * 16;
  const int arow = lane & 15;
  const int koff = (lane >> 4) * 2;
  const int col  = lane & 15;

  __shared__ float sA[16 * SROW_S];
  for (int i = tid; i < 16 * D_DIM; i += 256) {
    const int r = i >> 8, k = i & (D_DIM - 1);
    const float dc = fminf(fmaxf(dvec[k], 0.0f), 1.0f);
    sA[r * SROW_S + k] = w[(size_t)(i0 + r) * D_DIM + k] * dc;
  }
  __syncthreads();

  v8f acc = {0.f, 0.f, 0.f, 0.f, 0.f, 0.f, 0.f, 0.f};
  #pragma unroll 4
  for (int k = 0; k < D_DIM; k += 4) {
    v2f a, b;
    a.x = sA[arow * SROW_S + k + koff];
    a.y = sA[arow * SROW_S + k + koff + 1];
    b.x = w[(size_t)(jw + col) * D_DIM + k + koff];          // B[k][j]=w[j][k]
    b.y = w[(size_t)(jw + col) * D_DIM + k + koff + 1];
    acc = wmma_f32_16x16x4(a, b, acc);
  }

  const int rbase = (lane >> 4) * 8;
  #pragma unroll
  for (int v = 0; v < 8; ++v) {
    weff[(size_t)(i0 + rbase + v) * D_DIM + jw + col] = acc[v];
  }
}

// ---------------------------------------------------------------------------
// Kernel 3: fused Euler step. 2x2 register-blocked WMMA, TDM double-buffered
// software pipeline. Block = 8 waves covering 32 n-rows x 256 d-cols.
//   part 1: accA = adj[n0:n0+32,:] @ h[b]   (K=1024, 32 chunks, TDM pipelined)
//   part 2: accW = h[b,n0:n0+32,:] @ w_eff  (runs inside chunk m0==n0, whose
//            resident tile IS the part-2 A-tile -> no extra load)
//   out = h + dt*( sig(alpha)*0.5*(accA - h) + accW - h + x0 )
// LDS: 2*(32*260 + 32*33)*4 = 75,008 B (<= 320 KB/WG on CDNA5, 4 WG/WGP)
// ---------------------------------------------------------------------------
__global__ void __launch_bounds__(256)
step_kernel(const float* __restrict__ hin, const float* __restrict__ adj,
            const float* __restrict__ alpha, const float* __restrict__ x0,
            const float* __restrict__ weff, float* __restrict__ hout)
{
  const int b    = blockIdx.x;
  const int n0   = blockIdx.y * 32;
  const int tid  = threadIdx.x;
  const int wave = tid >> 5;
  const int lane = tid & 31;
  const int dw   = wave * 32;          // 8 waves * 32 cols = 256 = D
  const int arow = lane & 15;
  const int koff = (lane >> 4) * 2;
  const int col  = lane & 31 & 15;

  __shared__ float sH[2][KCH * SH_S];     // 2 x (32 x 260) padded h tiles
  __shared__ float sAdj[2][32 * SADJ_S];  // 2 x (32 x 33)  padded adj tiles

  const float* __restrict__ hb = hin + (size_t)b * N_DIM * D_DIM;
  const int cstar = n0 / KCH;             // chunk whose rows == part-2 A rows

  const v8f zv = {0.f, 0.f, 0.f, 0.f, 0.f, 0.f, 0.f, 0.f};
  v8f accA[2][2] = {{zv, zv}, {zv, zv}};  // [row-half][col-half] adj @ h
  v8f accW[2][2] = {{zv, zv}, {zv, zv}};  // h @ w_eff

  // --- per-chunk WMMA compute (part 1) ---
  auto compute_chunk = [&](const float* sHb, const float* sAb) {
    #pragma unroll 2
    for (int k = 0; k < KCH; k += 4) {
      v2f a0, a1, b0, b1;
      a0.x = sAb[arow * SADJ_S + k + koff];
      a0.y = sAb[arow * SADJ_S + k + koff + 1];
      a1.x = sAb[(arow + 16) * SADJ_S + k + koff];
      a1.y = sAb[(arow + 16) * SADJ_S + k + koff + 1];
      b0.x = sHb[(k + koff) * SH_S + dw + col];
      b0.y = sHb[(k + koff + 1) * SH_S + dw + col];
      b1.x = sHb[(k + koff) * SH_S + dw + 16 + col];
      b1.y = sHb[(k + koff + 1) * SH_S + dw + 16 + col];
      accA[0][0] = wmma_f32_16x16x4(a0, b0, accA[0][0]);
      accA[0][1] = wmma_f32_16x16x4(a0, b1, accA[0][1]);
      accA[1][0] = wmma_f32_16x16x4(a1, b0, accA[1][0]);
      accA[1][1] = wmma_f32_16x16x4(a1, b1, accA[1][1]);
    }
  };
  // --- part-2 WMMA compute: A = resident h-row tile, B = w_eff from L2 ---
  auto compute_w = [&](const float* sHb) {
    #pragma unroll 2
    for (int k = 0; k < D_DIM; k += 4) {
      v2f a0, a1, b0, b1;
      a0.x = sHb[arow * SH_S + k + koff];
      a0.y = sHb[arow * SH_S + k + koff + 1];
      a1.x = sHb[(arow + 16) * SH_S + k + koff];
      a1.y = sHb[(arow + 16) * SH_S + k + koff + 1];
      b0.x = weff[(size_t)(k + koff) * D_DIM + dw + col];
      b0.y = weff[(size_t)(k + koff + 1) * D_DIM + dw + col];
      b1.x = weff[(size_t)(k + koff) * D_DIM + dw + 16 + col];
      b1.y = weff[(size_t)(k + koff + 1) * D_DIM + dw + 16 + col];
      accW[0][0] = wmma_f32_16x16x4(a0, b0, accW[0][0]);
      accW[0][1] = wmma_f32_16x16x4(a0, b1, accW[0][1]);
      accW[1][0] = wmma_f32_16x16x4(a1, b0, accW[1][0]);
      accW[1][1] = wmma_f32_16x16x4(a1, b1, accW[1][1]);
    }
  };

#if HAVE_TDM
  auto issue_chunk = [&](int c, int buf) {
    const int m0 = c * KCH;
    // h[b, m0:m0+32, 0:256] -> sH[buf]  (TDM pads 4 dw / 256 dw -> stride 260)
    tdm_load_2d_f32((unsigned)(uintptr_t)&sH[buf][0], hb + (size_t)m0 * D_DIM,
                    D_DIM, KCH, D_DIM, N_DIM, D_DIM, /*256 dw*/ 7, /*4 dw*/ 3);
    // adj[n0:n0+32, m0:m0+32] -> sAdj[buf] (pads 1 dw / 32 dw -> stride 33)
    tdm_load_2d_f32((unsigned)(uintptr_t)&sAdj[buf][0],
                    adj + (size_t)n0 * N_DIM + m0,
                    KCH, 32, N_DIM, N_DIM, N_DIM, /*32 dw*/ 4, /*1 dw*/ 0);
  };

  // prologue: put chunks 0 and 1 in flight; wait until chunk 0 has landed
  // (TDM ops of one wave complete in order -> tensorcnt<=2 means chunk0 done)
  if (wave == 0) {
    issue_chunk(0, 0);
    issue_chunk(1, 1);
    __builtin_amdgcn_s_wait_tensorcnt(2);
  }
  __syncthreads();

  for (int c = 0; c < NCHUNK; ++c) {
    const int cur = c & 1;
    compute_chunk(&sH[cur][0], &sAdj[cur][0]);
    if (c == cstar) compute_w(&sH[cur][0]);   // tile == part-2 A-tile: fuse
    __syncthreads();                          // all waves done reading buf cur
    if (wave == 0) {
      if (c + 2 < NCHUNK) {
        issue_chunk(c + 2, cur);              // refill just-freed buffer
        __builtin_amdgcn_s_wait_tensorcnt(2); // chunk c+1 landed
      } else if (c + 1 < NCHUNK) {
        __builtin_amdgcn_s_wait_tensorcnt(0); // last chunk landed
      }
    }
    __syncthreads();                          // publish next buffer
  }
#else
  for (int c = 0; c < NCHUNK; ++c) {
    const int cur = c & 1;
    const int m0 = c * KCH;
    for (int i = tid; i < KCH * (D_DIM / 4); i += 256) {
      const int r = i >> 6, c4 = i & 63;
      *(float4*)&sH[cur][r * SH_S + c4 * 4] =
          *(const float4*)(hb + (size_t)(m0 + r) * D_DIM + c4 * 4);
    }
    for (int i = tid; i < 32 * KCH; i += 256) {
      const int r = i >> 5, cc = i & 31;
      sAdj[cur][r * SADJ_S + cc] = adj[(size_t)(n0 + r) * N_DIM + m0 + cc];
    }
    __syncthreads();
    compute_chunk(&sH[cur][0], &sAdj[cur][0]);
    if (c == cstar) compute_w(&sH[cur][0]);
    __syncthreads();
  }
#endif

  // ---- epilogue: fused Euler update ----------------------------------------
  const int rbase = (lane >> 4) * 8;
  #pragma unroll
  for (int rh = 0; rh < 2; ++rh) {
    #pragma unroll
    for (int v = 0; v < 8; ++v) {
      const int n = n0 + rh * 16 + rbase + v;
      const float al = 1.0f / (1.0f + __expf(-alpha[n]));
      #pragma unroll
      for (int ch = 0; ch < 2; ++ch) {
        const size_t idx =
            (size_t)b * N_DIM * D_DIM + (size_t)n * D_DIM + dw + ch * 16 + col;
        const float hold = hin[idx];
        const float f = al * 0.5f * (accA[rh][ch][v] - hold) +
                        accW[rh][ch][v] - hold + x0[idx];
        hout[idx] = hold + DT * f;
      }
    }
  }
}

// ---------------------------------------------------------------------------
extern "C" void kernel_launch(void* const* d_in, const int* in_sizes, int n_in,
                              void* d_out, int out_size, void* d_ws, size_t ws_size,
                              hipStream_t stream) {
  const float* x     = (const float*)d_in[0];  // [B, N, D]
  const float* adj   = (const float*)d_in[1];  // [N, N]
  const float* alpha = (const float*)d_in[2];  // [N]
  const float* w     = (const float*)d_in[3];  // [D, D]
  const float* dvec  = (const float*)d_in[4];  // [D]
  const float* fcw   = (const float*)d_in[5];  // [D, D]
  const float* fcb   = (const float*)d_in[6];  // [D]
  // d_in[7] = vt (unused)

  float* hA   = (float*)d_out;                                 // 32 MB
  float* hB   = (float*)d_ws;                                  // 32 MB
  float* weff = (float*)d_ws + (size_t)B_DIM * N_DIM * D_DIM;  // 256 KB

  const dim3 blk(256);

  fcin_kernel<<<dim3(B_DIM * N_DIM / 16, D_DIM / 128), blk, 0, stream>>>(
      x, fcw, fcb, hA);
  weff_kernel<<<dim3(D_DIM / 16, D_DIM / 128), blk, 0, stream>>>(
      w, dvec, weff);

  // 8 Euler steps, ping-pong; s=7 writes hA == d_out
  const float* cur = hA;
  float* nxt = hB;
  for (int s = 0; s < STEPS; ++s) {
    step_kernel<<<dim3(B_DIM, N_DIM / 32, 1), blk, 0, stream>>>(
        cur, adj, alpha, x, weff, nxt);
    float* t = (float*)cur;
    cur = nxt;
    nxt = t;
  }
}